// CSConv2D_73057393705079
// MI455X (gfx1250) — compile-verified
//
#include <hip/hip_runtime.h>

// Problem constants (fixed by the reference)
#define KSZ      5
#define NBUCKETS 25
#define BB       8
#define CC       96
#define HH       192
#define WW       192

// Tiling
#define TLH      16                // tile height (pixels)
#define TLW      32                // tile width  (pixels)
#define HALO     4                 // (KSZ-1)
#define LH       (TLH + HALO)      // 20 halo rows
#define LW       (TLW + HALO)      // 36 halo cols
#define LSTRIDE  38                // padded LDS row stride (floats) - kills bank conflicts
#define NELEM    (LH * LW)         // 720 halo elements per channel
#define CGROUPS  2                 // split channels across blocks for occupancy
#define CPG      (CC / CGROUPS)    // 48 channels per block
#define NTHREADS 256
#define NISSUE   3                 // ceil(NELEM / NTHREADS)

typedef float f32x2 __attribute__((ext_vector_type(2)));

#if defined(__HIP_DEVICE_COMPILE__) && __has_builtin(__builtin_amdgcn_global_load_async_to_lds_b32)
#define USE_ASYNC 1
typedef __attribute__((address_space(1))) int g1_i32;   // global (AS1) int
typedef __attribute__((address_space(3))) int l3_i32;   // LDS (AS3) int
#else
#define USE_ASYNC 0
#endif

__device__ __forceinline__ void wait_async_zero() {
#if USE_ASYNC
#if __has_builtin(__builtin_amdgcn_s_wait_asynccnt)
    __builtin_amdgcn_s_wait_asynccnt(0);
#else
    asm volatile("s_wait_asynccnt 0" ::: "memory");
#endif
#endif
}

__global__ __launch_bounds__(NTHREADS) void csconv2d_async_kernel(
    const float* __restrict__ input,     // [B,C,H,W]
    const float* __restrict__ bank,      // [25,5,5]
    const int*   __restrict__ buckets,   // [B,H,W]
    float*       __restrict__ out)       // [B,C,H,W]
{
    __shared__ float sbank[NBUCKETS * KSZ * KSZ];   // 2.5 KB
    __shared__ float sbuf[2][LH * LSTRIDE];         // double-buffered halo tile

    const int tid   = threadIdx.x;
    const int tileX = blockIdx.x;            // 0..5
    const int tileY = blockIdx.y;            // 0..11
    const int bz    = blockIdx.z;            // 0..15
    const int b     = bz / CGROUPS;
    const int c0    = (bz % CGROUPS) * CPG;

    const int gh0 = tileY * TLH;
    const int gw0 = tileX * TLW;

    // Stage kernel bank into LDS; zero BOTH halo buffers once.
    // OOB halo cells are never written by the async loads, so these zeros
    // implement the 'same' zero padding for every channel iteration.
    for (int e = tid; e < NBUCKETS * KSZ * KSZ; e += NTHREADS) sbank[e] = bank[e];
    {
        float* z = &sbuf[0][0];
        for (int e = tid; e < 2 * LH * LSTRIDE; e += NTHREADS) z[e] = 0.0f;
    }
    __syncthreads();

    // This thread owns output pixels (gh, gw) and (gh, gw+1).
    const int ty = tid >> 4;                 // 0..15
    const int tx = tid & 15;                 // 0..15
    const int gh = gh0 + ty;
    const int gw = gw0 + 2 * tx;

    // Gather per-pixel 25-tap weight vectors once; reuse across 48 channels.
    const int bk0 = buckets[((size_t)b * HH + gh) * WW + gw] * (KSZ * KSZ);
    const int bk1 = buckets[((size_t)b * HH + gh) * WW + gw + 1] * (KSZ * KSZ);
    float w0[25], w1[25];
#pragma unroll
    for (int t = 0; t < 25; ++t) {
        w0[t] = sbank[bk0 + t];
        w1[t] = sbank[bk1 + t];
    }

    // Channel-invariant halo addressing: precompute this thread's (up to 3)
    // staging slots once; per channel only the base pointer changes.
    unsigned ldsoff[NISSUE];                 // byte offset within one sbuf buffer
    unsigned goff[NISSUE];                   // byte offset within one channel plane
    bool     valid[NISSUE];
#pragma unroll
    for (int k = 0; k < NISSUE; ++k) {
        const int e   = tid + k * NTHREADS;
        const int r   = e / LW;
        const int col = e - r * LW;
        const int sy  = gh0 - 2 + r;
        const int sx  = gw0 - 2 + col;
        valid[k]  = (e < NELEM) && ((unsigned)sy < (unsigned)HH) &&
                    ((unsigned)sx < (unsigned)WW);
        ldsoff[k] = (unsigned)(r * LSTRIDE + col) * 4u;
        goff[k]   = (unsigned)(sy * WW + sx) * 4u;   // unused when !valid
    }

    // Issue the halo-tile staging for channel c into buffer `sel`.
    auto issue = [&](int c, int sel) {
        const char* srcb = (const char*)input +
                           (size_t)(b * CC + c) * (HH * WW) * sizeof(float);
        char* ldsb = (char*)&sbuf[sel][0];
#pragma unroll
        for (int k = 0; k < NISSUE; ++k) {
            if (valid[k]) {
#if USE_ASYNC
                __builtin_amdgcn_global_load_async_to_lds_b32(
                    (g1_i32*)(void*)(srcb + goff[k]),
                    (l3_i32*)(void*)(ldsb + ldsoff[k]), 0, 0);
#else
                *(float*)(ldsb + ldsoff[k]) = *(const float*)(srcb + goff[k]);
#endif
            }
        }
    };

    int sel = 0;
    issue(c0, 0);                            // prefetch first channel

    for (int cc = 0; cc < CPG; ++cc) {
        const int c = c0 + cc;

        wait_async_zero();                   // this wave's async writes landed
        __syncthreads();                     // => all waves' writes visible

        if (cc + 1 < CPG) issue(c + 1, sel ^ 1);  // overlap next staging with compute

        const float* bp = &sbuf[sel][0];
        float acc0 = 0.0f, acc1 = 0.0f;
#pragma unroll
        for (int i = 0; i < KSZ; ++i) {
            const float* row = bp + (ty + i) * LSTRIDE + 2 * tx;
            const float v0 = row[0], v1 = row[1], v2 = row[2];
            const float v3 = row[3], v4 = row[4], v5 = row[5];
            acc0 = fmaf(w0[5 * i + 0], v0, acc0);
            acc0 = fmaf(w0[5 * i + 1], v1, acc0);
            acc0 = fmaf(w0[5 * i + 2], v2, acc0);
            acc0 = fmaf(w0[5 * i + 3], v3, acc0);
            acc0 = fmaf(w0[5 * i + 4], v4, acc0);
            acc1 = fmaf(w1[5 * i + 0], v1, acc1);
            acc1 = fmaf(w1[5 * i + 1], v2, acc1);
            acc1 = fmaf(w1[5 * i + 2], v3, acc1);
            acc1 = fmaf(w1[5 * i + 3], v4, acc1);
            acc1 = fmaf(w1[5 * i + 4], v5, acc1);
        }

        // Streaming (non-temporal) 8B store: output is write-once, keep it
        // from evicting reusable input halo lines in L2.
        const size_t oidx = ((size_t)(b * CC + c) * HH + gh) * WW + gw;
        f32x2 o; o.x = acc0; o.y = acc1;     // 8B aligned: gw is even
        __builtin_nontemporal_store(o, (f32x2*)(out + oidx));

        sel ^= 1;
    }
}

extern "C" void kernel_launch(void* const* d_in, const int* in_sizes, int n_in,
                              void* d_out, int out_size, void* d_ws, size_t ws_size,
                              hipStream_t stream) {
    (void)in_sizes; (void)n_in; (void)out_size; (void)d_ws; (void)ws_size;
    const float* input   = (const float*)d_in[0];
    const float* bank    = (const float*)d_in[1];
    const int*   buckets = (const int*)d_in[2];
    float*       out     = (float*)d_out;

    dim3 grid(WW / TLW, HH / TLH, BB * CGROUPS);   // 6 x 12 x 16 = 1152 blocks
    dim3 block(NTHREADS);
    hipLaunchKernelGGL(csconv2d_async_kernel, grid, block, 0, stream,
                       input, bank, buckets, out);
}